// Edge_cycle_51531017617888
// MI455X (gfx1250) — compile-verified
//
#include <hip/hip_runtime.h>

// Problem constants (match reference)
#define DREP 64
#define SCYC 6
#define NEDGE 400000
#define NCYC 150000
#define NROWS (NCYC * SCYC)   // 900000
#define NMSG  (2 * NROWS)     // 1800000
#define NHID  128
#define BN_EPS 1e-5f

typedef __attribute__((ext_vector_type(2))) float v2f;
typedef __attribute__((ext_vector_type(8))) float v8f;

// ---------------------------------------------------------------- utilities

__global__ void zero_kernel(float* __restrict__ p, long n) {
    long i = (long)blockIdx.x * blockDim.x + threadIdx.x;
    long stride = (long)gridDim.x * blockDim.x;
    for (; i < n; i += stride) p[i] = 0.0f;
}

// segment_sum(src[sidx], didx): one thread per (message, dim) pair, D=64 dims
__global__ void scatter_add_kernel(const float* __restrict__ src,
                                   const int* __restrict__ sidx,
                                   const int* __restrict__ didx,
                                   float* __restrict__ acc, int nmsg) {
    long i = (long)blockIdx.x * blockDim.x + threadIdx.x;
    long total = (long)nmsg * DREP;
    long stride = (long)gridDim.x * blockDim.x;
    for (; i < total; i += stride) {
        int m = (int)(i >> 6);
        int d = (int)(i & 63);
        float v = src[(long)sidx[m] * DREP + d];
        atomicAdd(&acc[(long)didx[m] * DREP + d], v);
    }
}

// csum[c][d] = sum_{s<6} e2c[c*6+s][d]
__global__ void cycle_sum_kernel(const float* __restrict__ e2c,
                                 float* __restrict__ csum) {
    long i = (long)blockIdx.x * blockDim.x + threadIdx.x;
    long total = (long)NCYC * DREP;
    long stride = (long)gridDim.x * blockDim.x;
    for (; i < total; i += stride) {
        int c = (int)(i >> 6);
        int d = (int)(i & 63);
        const float* p = e2c + (long)c * SCYC * DREP + d;
        float s = 0.0f;
#pragma unroll
        for (int j = 0; j < SCYC; ++j) s += p[j * DREP];
        csum[i] = s;
    }
}

// ------------------------------------------------- fused-A WMMA fp32 GEMM
// MODE 0: A = A0 [Mrows x K] row-major
// MODE 1: cycle-MLP layer1, K=320: [0:64)=cycle_rep(A0), [64:192)=e2c_lin(A1),
//         [192:256)=e2c(A2), [256:320)=csum[r/6](A3)
// MODE 2: edge-MLP layer1, K=128: [0:64)=edge_rep(A0), [64:128)=c2e(A1)
// Returns the address of element (r, k); (k, k+1) are always contiguous within
// one region because regions start at multiples of 64 and k is even.
template <int MODE>
__device__ __forceinline__ const float* addrA(const float* __restrict__ A0,
                                              const float* __restrict__ A1,
                                              const float* __restrict__ A2,
                                              const float* __restrict__ A3,
                                              int r, int k, int K) {
    if constexpr (MODE == 0) {
        return A0 + (long)r * K + k;
    } else if constexpr (MODE == 1) {
        if (k < 64)       return A0 + (long)r * 64 + k;
        else if (k < 192) return A1 + (long)r * 128 + (k - 64);
        else if (k < 256) return A2 + (long)r * 64 + (k - 192);
        else              return A3 + (long)(r / SCYC) * 64 + (k - 256);
    } else {
        if (k < 64) return A0 + (long)r * 64 + k;
        else        return A1 + (long)r * 64 + (k - 64);
    }
}

// Block = 256 threads = 8 wave32; each wave owns a 16-row strip.
// Block tile = 128 rows x N (N = NT*16, NT=4 or 8). B chunked 128 K-rows in
// LDS, stored pair-interleaved: ldsB[p*2N + 2n] = {B[kc+2p][n], B[kc+2p+1][n]}
// so a B fragment is one aligned ds_load_b64.
// Out-of-range rows are CLAMPED (not zeroed): WMMA rows are independent and
// the stores are guarded, so clamping keeps every A load unconditional.
template <int MODE, int NT>
__global__ __launch_bounds__(256) void gemm_wmma_f32(
    const float* __restrict__ A0, const float* __restrict__ A1,
    const float* __restrict__ A2, const float* __restrict__ A3,
    const float* __restrict__ Bm, const float* __restrict__ bias,
    float* __restrict__ out, int Mrows, int K) {
    constexpr int N = NT * 16;
    constexpr int NSHIFT = (N == 128) ? 7 : 6;
    __shared__ __align__(16) float ldsB[128 * N];   // <= 64 KB

    const int tid  = threadIdx.x;
    const int wave = tid >> 5;
    const int lane = tid & 31;
    const int half = lane >> 4;      // 0: K pair (0,1), 1: K pair (2,3)
    const int l16  = lane & 15;
    const int r0   = blockIdx.x * 128 + wave * 16;
    const int mrow = r0 + l16;
    const int rcl  = (mrow < Mrows) ? mrow : (Mrows - 1);   // clamp, not zero

    v8f acc[NT];
#pragma unroll
    for (int t = 0; t < NT; ++t)
#pragma unroll
        for (int v = 0; v < 8; ++v) acc[t][v] = 0.0f;

    for (int kc = 0; kc < K; kc += 128) {
        const int kcend = (kc + 128 < K) ? (kc + 128) : K;
        const int klen  = kcend - kc;          // always even
        __syncthreads();
        // Stage B chunk pair-interleaved
        for (int idx = tid; idx < (klen >> 1) * N; idx += 256) {
            const int p = idx >> NSHIFT;
            const int n = idx & (N - 1);
            v2f w;
            w.x = Bm[(long)(kc + 2 * p) * N + n];
            w.y = Bm[(long)(kc + 2 * p + 1) * N + n];
            *reinterpret_cast<v2f*>(&ldsB[p * 2 * N + 2 * n]) = w;
        }
        __syncthreads();

        // Prefetch this lane's A line for the next K chunk
        if (kcend < K)
            __builtin_prefetch(addrA<MODE>(A0, A1, A2, A3, rcl, kcend + 2 * half, K), 0, 1);

#pragma unroll 2
        for (int k0 = kc; k0 < kcend; k0 += 4) {
            const int kb = k0 + 2 * half;                 // per-half-wave K pair
            const v2f a = *reinterpret_cast<const v2f*>(
                addrA<MODE>(A0, A1, A2, A3, rcl, kb, K)); // one global_load_b64
            const int pb = ((k0 - kc) >> 1) + half;       // pair index in LDS
#pragma unroll
            for (int t = 0; t < NT; ++t) {
                const v2f b = *reinterpret_cast<const v2f*>(
                    &ldsB[pb * 2 * N + 2 * (t * 16 + l16)]); // one ds_load_b64
                acc[t] = __builtin_amdgcn_wmma_f32_16x16x4_f32(
                    false, a, false, b, (short)0, acc[t], false, false);
            }
        }
    }

    // C/D layout: vgpr v, lanes 0-15 -> row r0+v, lanes 16-31 -> row r0+v+8
#pragma unroll
    for (int t = 0; t < NT; ++t) {
        const int col = t * 16 + l16;
        const float bv = bias[col];
#pragma unroll
        for (int v = 0; v < 8; ++v) {
            const int row = r0 + v + 8 * half;
            if (row < Mrows) out[(long)row * N + col] = acc[t][v] + bv;
        }
    }
}

// ------------------------------------------------------------- BatchNorm
// Column sums and sum-of-squares over Mrows (N fixed at 128)
__global__ void bn_stats_kernel(const float* __restrict__ t,
                                float* __restrict__ stats, int Mrows) {
    const int col = threadIdx.x;       // blockDim.x == 128
    float s1 = 0.0f, s2 = 0.0f;
    for (int r = blockIdx.x; r < Mrows; r += gridDim.x) {
        float v = t[(long)r * NHID + col];
        s1 += v;
        s2 += v * v;
    }
    atomicAdd(&stats[col], s1);
    atomicAdd(&stats[NHID + col], s2);
}

// In-place t = relu(g*(t-mean)*rsqrt(var+eps)+b), population var
__global__ void bn_apply_relu_kernel(float* __restrict__ t,
                                     const float* __restrict__ stats,
                                     const float* __restrict__ g,
                                     const float* __restrict__ b, int Mrows) {
    long i = (long)blockIdx.x * blockDim.x + threadIdx.x;
    long total = (long)Mrows * NHID;
    long stride = (long)gridDim.x * blockDim.x;
    const float inv_m = 1.0f / (float)Mrows;
    for (; i < total; i += stride) {
        const int col = (int)(i & (NHID - 1));
        const float mean = stats[col] * inv_m;
        const float var  = stats[NHID + col] * inv_m - mean * mean;
        const float x = t[i];
        const float y = g[col] * (x - mean) * rsqrtf(var + BN_EPS) + b[col];
        t[i] = y > 0.0f ? y : 0.0f;
    }
}

// ---------------------------------------------------------------- launch

extern "C" void kernel_launch(void* const* d_in, const int* in_sizes, int n_in,
                              void* d_out, int out_size, void* d_ws, size_t ws_size,
                              hipStream_t stream) {
    const float* edge_rep  = (const float*)d_in[0];
    const float* cycle_rep = (const float*)d_in[1];
    const int*   e2c_src   = (const int*)d_in[2];
    const int*   e2c_dst   = (const int*)d_in[3];
    const int*   c2e_src   = (const int*)d_in[4];
    const int*   c2e_dst   = (const int*)d_in[5];
    const float* Wc  = (const float*)d_in[6];
    const float* bc  = (const float*)d_in[7];
    const float* cw1 = (const float*)d_in[8];
    const float* cb1 = (const float*)d_in[9];
    const float* cg1 = (const float*)d_in[10];
    const float* cB1 = (const float*)d_in[11];
    const float* cw2 = (const float*)d_in[12];
    const float* cb2 = (const float*)d_in[13];
    const float* cg2 = (const float*)d_in[14];
    const float* cB2 = (const float*)d_in[15];
    const float* cw3 = (const float*)d_in[16];
    const float* cb3 = (const float*)d_in[17];
    const float* ew1 = (const float*)d_in[18];
    const float* eb1 = (const float*)d_in[19];
    const float* eg1 = (const float*)d_in[20];
    const float* eB1 = (const float*)d_in[21];
    const float* ew2 = (const float*)d_in[22];
    const float* eb2 = (const float*)d_in[23];

    float* out       = (float*)d_out;
    float* edge_out  = out;                         // [E, 64]
    float* cycle_out = out + (size_t)NEDGE * DREP;  // [ROWS, 64]

    // Workspace layout (~1.96 GB of fp32)
    float* ws     = (float*)d_ws;
    float* e2c    = ws;                               // ROWS*64
    float* csum   = e2c    + (size_t)NROWS * DREP;    // C*64
    float* e2clin = csum   + (size_t)NCYC  * DREP;    // ROWS*128
    float* t1     = e2clin + (size_t)NROWS * NHID;    // ROWS*128 (h1 in-place)
    float* t2     = t1     + (size_t)NROWS * NHID;    // ROWS*128 (h2 in-place)
    float* c2e    = t2     + (size_t)NROWS * NHID;    // E*64
    float* t3     = c2e    + (size_t)NEDGE * DREP;    // E*128 (h3 in-place)
    float* stats  = t3     + (size_t)NEDGE * NHID;    // 3 x 256
    float* stats0 = stats;
    float* stats1 = stats + 256;
    float* stats2 = stats + 512;

    const dim3 tb(256);
    const dim3 zgrid(2048);
    const dim3 sgrid(8192);
    const dim3 statgrid(2048);
    const dim3 cyc_gemm((NROWS + 127) / 128);   // 7032 blocks
    const dim3 edg_gemm((NEDGE + 127) / 128);   // 3125 blocks

    // Zero accumulators + stats (ws is poisoned; must re-zero every call)
    zero_kernel<<<zgrid, tb, 0, stream>>>(e2c, (long)NROWS * DREP);
    zero_kernel<<<zgrid, tb, 0, stream>>>(c2e, (long)NEDGE * DREP);
    zero_kernel<<<dim3(4), tb, 0, stream>>>(stats, 768);

    // e2c = segment_sum(edge_rep[e2c_src], e2c_dst)
    scatter_add_kernel<<<sgrid, tb, 0, stream>>>(edge_rep, e2c_src, e2c_dst, e2c, NMSG);

    // e2c_lin = e2c @ Wc + bc          [ROWS,64]x[64,128]
    gemm_wmma_f32<0, 8><<<cyc_gemm, tb, 0, stream>>>(
        e2c, nullptr, nullptr, nullptr, Wc, bc, e2clin, NROWS, DREP);

    // per-cycle sums for the broadcast part of linmap
    cycle_sum_kernel<<<sgrid, tb, 0, stream>>>(e2c, csum);

    // t1 = concat(cycle_rep, e2c_lin, e2c, csum-bcast) @ cw1 + cb1   K=320
    gemm_wmma_f32<1, 8><<<cyc_gemm, tb, 0, stream>>>(
        cycle_rep, e2clin, e2c, csum, cw1, cb1, t1, NROWS, 320);
    bn_stats_kernel<<<statgrid, dim3(NHID), 0, stream>>>(t1, stats0, NROWS);
    bn_apply_relu_kernel<<<sgrid, tb, 0, stream>>>(t1, stats0, cg1, cB1, NROWS);

    // t2 = h1 @ cw2 + cb2              [ROWS,128]x[128,128]
    gemm_wmma_f32<0, 8><<<cyc_gemm, tb, 0, stream>>>(
        t1, nullptr, nullptr, nullptr, cw2, cb2, t2, NROWS, NHID);
    bn_stats_kernel<<<statgrid, dim3(NHID), 0, stream>>>(t2, stats1, NROWS);
    bn_apply_relu_kernel<<<sgrid, tb, 0, stream>>>(t2, stats1, cg2, cB2, NROWS);

    // cycle_out = h2 @ cw3 + cb3       [ROWS,128]x[128,64] -> d_out segment 2
    gemm_wmma_f32<0, 4><<<cyc_gemm, tb, 0, stream>>>(
        t2, nullptr, nullptr, nullptr, cw3, cb3, cycle_out, NROWS, NHID);

    // c2e = segment_sum(cycle_out[c2e_src], c2e_dst)
    scatter_add_kernel<<<sgrid, tb, 0, stream>>>(cycle_out, c2e_src, c2e_dst, c2e, NMSG);

    // t3 = concat(edge_rep, c2e) @ ew1 + eb1   K=128
    gemm_wmma_f32<2, 8><<<edg_gemm, tb, 0, stream>>>(
        edge_rep, c2e, nullptr, nullptr, ew1, eb1, t3, NEDGE, NHID);
    bn_stats_kernel<<<statgrid, dim3(NHID), 0, stream>>>(t3, stats2, NEDGE);
    bn_apply_relu_kernel<<<sgrid, tb, 0, stream>>>(t3, stats2, eg1, eB1, NEDGE);

    // edge_out = h3 @ ew2 + eb2        [E,128]x[128,64] -> d_out segment 1
    gemm_wmma_f32<0, 4><<<edg_gemm, tb, 0, stream>>>(
        t3, nullptr, nullptr, nullptr, ew2, eb2, edge_out, NEDGE, NHID);
}